// CBAMChannelAttn_42279658062007
// MI455X (gfx1250) — compile-verified
//
#include <hip/hip_runtime.h>
#include <hip/hip_bf16.h>
#include <math.h>

// ---------------------------------------------------------------------------
// CBAM channel attention, MI455X (gfx1250, wave32).
//   x:[32,768,56,56] f32, w1:[48,768], w2:[768,48]
//   gate = sigmoid( mlp(avg) + mlp(max) ),  out = x * gate[b,c]
// Bandwidth-bound: ~924 MB traffic => ~40us @ 23.3 TB/s. MLP done with
// V_WMMA_F32_16X16X4_F32 (fp32 in/out, exact precision, tiny GEMMs).
// ---------------------------------------------------------------------------

typedef __attribute__((ext_vector_type(2))) float v2f;
typedef __attribute__((ext_vector_type(8))) float v8f;

#define B 32
#define C 768
#define HW 3136          // 56*56
#define HW4 784          // HW/4
#define HID 48           // 768/16
#define NPLANE (B * C)   // 24576

// ---------------------------------------------------------------------------
// Kernel 1: per-(b,c) spatial avg+max pooling.
// pooled layout: [64][768] row-major. Rows 0..31 = avg, rows 32..63 = max.
// ---------------------------------------------------------------------------
__global__ void cbam_pool(const float* __restrict__ x, float* __restrict__ pooled) {
    const int p = blockIdx.x;                       // p = b*768 + c
    const float4* xp = (const float4*)(x + (size_t)p * HW);

    float s = 0.0f;
    float m = -INFINITY;
    for (int i = threadIdx.x; i < HW4; i += blockDim.x) {
        float4 v = xp[i];
        s += (v.x + v.y) + (v.z + v.w);
        m = fmaxf(m, fmaxf(fmaxf(v.x, v.y), fmaxf(v.z, v.w)));
    }
    // wave32 reduction
    #pragma unroll
    for (int o = 16; o > 0; o >>= 1) {
        s += __shfl_down(s, o, 32);
        m = fmaxf(m, __shfl_down(m, o, 32));
    }
    __shared__ float ss[8], sm[8];
    const int wave = threadIdx.x >> 5;
    const int lane = threadIdx.x & 31;
    if (lane == 0) { ss[wave] = s; sm[wave] = m; }
    __syncthreads();
    if (threadIdx.x == 0) {
        const int nw = blockDim.x >> 5;
        float S = ss[0], M = sm[0];
        for (int w = 1; w < nw; ++w) { S += ss[w]; M = fmaxf(M, sm[w]); }
        pooled[p]              = S * (1.0f / (float)HW);   // avg rows 0..31
        pooled[B * C + p]      = M;                        // max rows 32..63
    }
}

// ---------------------------------------------------------------------------
// Kernel 2: h[64][48] = gelu( pooled[64][768] @ w1^T[768][48] )
// One 16x16 output tile per 32-thread block. K loop in steps of 4 with
// V_WMMA_F32_16X16X4_F32. Grid: 4 Mtiles x 3 Ntiles = 12 blocks.
// A frag (16x4 f32): lanes 0-15 hold K={0,1}, lanes 16-31 hold K={2,3}.
// B frag (4x16 f32): mirrored (lanes 0-15 rows {0,1}, lanes 16-31 rows {2,3}).
// C/D: VGPR v = rows m0+v (lanes 0-15) and m0+v+8 (lanes 16-31).
// ---------------------------------------------------------------------------
__global__ void cbam_mlp1(const float* __restrict__ pooled,
                          const float* __restrict__ w1,
                          float* __restrict__ h) {
    const int mt = blockIdx.x / 3;
    const int nt = blockIdx.x % 3;
    const int m0 = mt * 16, n0 = nt * 16;
    const int ln = threadIdx.x & 15;
    const int hi = (threadIdx.x >> 4) & 1;          // 0: K base 0, 1: K base 2

    const float* __restrict__ arow = pooled + (size_t)(m0 + ln) * C;
    const float* __restrict__ brow = w1     + (size_t)(n0 + ln) * C; // B[k][n]=w1[n*768+k]

    v8f acc = {};
    for (int k = 0; k < C; k += 4) {
        const int kk = k + hi * 2;
        v2f a = { arow[kk], arow[kk + 1] };
        v2f b = { brow[kk], brow[kk + 1] };
        acc = __builtin_amdgcn_wmma_f32_16x16x4_f32(
            false, a, false, b, (short)0, acc, false, false);
    }
    #pragma unroll
    for (int v = 0; v < 8; ++v) {
        float z = acc[v];
        float g = 0.5f * z * (1.0f + erff(z * 0.70710678118654752f)); // exact GELU
        h[(size_t)(m0 + v + hi * 8) * HID + (n0 + ln)] = g;
    }
}

// ---------------------------------------------------------------------------
// Kernel 3: gate[32][768] = sigmoid( (h_avg + h_max)[32][48] @ w2^T[48][768] )
// (stage-2 is linear, so pre-add the avg/max halves of h). Grid: 2 x 48 = 96
// one-wave blocks, K=48 -> 12 WMMA steps.
// ---------------------------------------------------------------------------
__global__ void cbam_mlp2(const float* __restrict__ h,
                          const float* __restrict__ w2,
                          float* __restrict__ gate) {
    const int mt = blockIdx.x / HID;                // 0..1
    const int nt = blockIdx.x % HID;                // 0..47
    const int m0 = mt * 16, n0 = nt * 16;
    const int ln = threadIdx.x & 15;
    const int hi = (threadIdx.x >> 4) & 1;

    const float* __restrict__ arow0 = h  + (size_t)(m0 + ln) * HID;       // avg half
    const float* __restrict__ arow1 = h  + (size_t)(m0 + ln + 32) * HID;  // max half
    const float* __restrict__ brow  = w2 + (size_t)(n0 + ln) * HID;       // B[k][c]=w2[c*48+k]

    v8f acc = {};
    #pragma unroll
    for (int k = 0; k < HID; k += 4) {
        const int kk = k + hi * 2;
        v2f a = { arow0[kk] + arow1[kk], arow0[kk + 1] + arow1[kk + 1] };
        v2f b = { brow[kk], brow[kk + 1] };
        acc = __builtin_amdgcn_wmma_f32_16x16x4_f32(
            false, a, false, b, (short)0, acc, false, false);
    }
    #pragma unroll
    for (int v = 0; v < 8; ++v) {
        float z = acc[v];
        gate[(size_t)(m0 + v + hi * 8) * C + (n0 + ln)] = 1.0f / (1.0f + expf(-z));
    }
}

// ---------------------------------------------------------------------------
// Kernel 4: out = x * gate[plane], one block per (b,c) plane, float4 streams.
// ---------------------------------------------------------------------------
__global__ void cbam_scale(const float* __restrict__ x,
                           const float* __restrict__ gate,
                           float* __restrict__ out) {
    const int p = blockIdx.x;
    const float g = gate[p];                        // uniform -> scalar load
    const float4* xp = (const float4*)(x + (size_t)p * HW);
    float4* op = (float4*)(out + (size_t)p * HW);
    for (int i = threadIdx.x; i < HW4; i += blockDim.x) {
        float4 v = xp[i];
        v.x *= g; v.y *= g; v.z *= g; v.w *= g;
        op[i] = v;
    }
}

// ---------------------------------------------------------------------------
// Launch. Workspace layout (floats):
//   pooled : [64*768]  = 49152   @ 0
//   h      : [64*48]   =  3072   @ 49152
//   gate   : [32*768]  = 24576   @ 52224        total 76800 f = 300 KiB
// ---------------------------------------------------------------------------
extern "C" void kernel_launch(void* const* d_in, const int* in_sizes, int n_in,
                              void* d_out, int out_size, void* d_ws, size_t ws_size,
                              hipStream_t stream) {
    const float* x  = (const float*)d_in[0];
    const float* w1 = (const float*)d_in[1];
    const float* w2 = (const float*)d_in[2];
    float* out = (float*)d_out;

    float* ws     = (float*)d_ws;
    float* pooled = ws;
    float* h      = ws + 64 * C;
    float* gate   = ws + 64 * C + 64 * HID;

    cbam_pool <<<NPLANE, 256, 0, stream>>>(x, pooled);
    cbam_mlp1 <<<12,     32,  0, stream>>>(pooled, w1, h);
    cbam_mlp2 <<<96,     32,  0, stream>>>(h, w2, gate);
    cbam_scale<<<NPLANE, 256, 0, stream>>>(x, gate, out);
}